// hash_embedding_73495480369708
// MI455X (gfx1250) — compile-verified
//
#include <hip/hip_runtime.h>
#include <math.h>

#define NLEV 16
#define BLOCK 256
#define PRIME1 2654435761u
#define PRIME2 805459861u
// Byte-offset domain: index mask (2^19-1) pre-shifted by 3 (8 bytes per float2 entry)
#define OFF_MASK 0x3FFFF8u
#define LVL_BYTES (size_t)(1u << 22)   // 2^19 entries * 8 bytes per level

typedef float v4f __attribute__((ext_vector_type(4)));

struct HalfRes { float h[NLEV]; };     // res[l] * 0.5, computed on host

__global__ __launch_bounds__(BLOCK) void hashgrid_embed_kernel(
    const float* __restrict__ x,
    const float* __restrict__ tables,
    float* __restrict__ out,
    HalfRes hres, int npts)
{
    __shared__ float s_x[BLOCK * 3];

    const int t = threadIdx.x;
    const int p = blockIdx.x * BLOCK + t;

    // ---- Stage this block's coordinates into LDS (CDNA5 async path) ----
    if ((blockIdx.x + 1) * BLOCK <= npts) {
        const float* gsrc = x + (size_t)blockIdx.x * (BLOCK * 3);
        const unsigned lbase = (unsigned)(size_t)(&s_x[0]);
#pragma unroll
        for (int k = 0; k < 3; ++k) {
            const unsigned loff = lbase + 4u * (unsigned)(t + BLOCK * k);
            const float* gp = gsrc + t + BLOCK * k;
            asm volatile("global_load_async_to_lds_b32 %0, %1, off"
                         :: "v"(loff), "v"(gp) : "memory");
        }
        asm volatile("s_wait_asynccnt 0" ::: "memory");
    } else {
        if (p < npts) {
            s_x[3 * t + 0] = x[3 * (size_t)p + 0];
            s_x[3 * t + 1] = x[3 * (size_t)p + 1];
            s_x[3 * t + 2] = x[3 * (size_t)p + 2];
        }
    }
    __syncthreads();

    if (p >= npts) return;

    const float px = s_x[3 * t + 0];
    const float py = s_x[3 * t + 1];
    const float pz = s_x[3 * t + 2];

    float2 o[NLEV];

#pragma unroll
    for (int l = 0; l < NLEV; ++l) {
        const float hr = hres.h[l];                 // res/2 == 1/grid
        // t = (x - BOX_MIN) / grid = (x+1) * res/2 ; bl = floor(t) ; w = t - bl
        const float tx = __fmaf_rn(px, hr, hr);
        const float ty = __fmaf_rn(py, hr, hr);
        const float tz = __fmaf_rn(pz, hr, hr);
        const float fx = floorf(tx);
        const float fy = floorf(ty);
        const float fz = floorf(tz);
        const float wx = tx - fx;
        const float wy = ty - fy;
        const float wz = tz - fz;

        // Hash, pre-scaled to byte offsets: (h & MASK) * 8 == (h*8) & (MASK*8),
        // and (b*P)<<3 == b*(P*8 mod 2^32).  (b+1)*P8 == b*P8 + P8.
        const unsigned hx0 = ((unsigned)(int)fx) << 3;
        const unsigned hx1 = hx0 + 8u;
        const unsigned hy0 = ((unsigned)(int)fy) * (PRIME1 * 8u);
        const unsigned hy1 = hy0 + (PRIME1 * 8u);
        const unsigned hz0 = ((unsigned)(int)fz) * (PRIME2 * 8u);
        const unsigned hz1 = hz0 + (PRIME2 * 8u);

        const unsigned hyz00 = hy0 ^ hz0;
        const unsigned hyz01 = hy0 ^ hz1;
        const unsigned hyz10 = hy1 ^ hz0;
        const unsigned hyz11 = hy1 ^ hz1;

        const char* __restrict__ tab = (const char*)tables + (size_t)l * LVL_BYTES;

        // Corner order matches BOX_OFFSETS: (i,j,k) = (x,y,z) offsets.
        const float2 v000 = *(const float2*)(tab + ((hx0 ^ hyz00) & OFF_MASK));
        const float2 v001 = *(const float2*)(tab + ((hx0 ^ hyz01) & OFF_MASK));
        const float2 v010 = *(const float2*)(tab + ((hx0 ^ hyz10) & OFF_MASK));
        const float2 v011 = *(const float2*)(tab + ((hx0 ^ hyz11) & OFF_MASK));
        const float2 v100 = *(const float2*)(tab + ((hx1 ^ hyz00) & OFF_MASK));
        const float2 v101 = *(const float2*)(tab + ((hx1 ^ hyz01) & OFF_MASK));
        const float2 v110 = *(const float2*)(tab + ((hx1 ^ hyz10) & OFF_MASK));
        const float2 v111 = *(const float2*)(tab + ((hx1 ^ hyz11) & OFF_MASK));

        // Trilinear: hoist (1-w) per axis; each lerp = 2 mul + 2 fma.
        const float ax = 1.0f - wx, ay = 1.0f - wy, az = 1.0f - wz;

        const float c00x = __fmaf_rn(v100.x, wx, v000.x * ax);
        const float c00y = __fmaf_rn(v100.y, wx, v000.y * ax);
        const float c01x = __fmaf_rn(v101.x, wx, v001.x * ax);
        const float c01y = __fmaf_rn(v101.y, wx, v001.y * ax);
        const float c10x = __fmaf_rn(v110.x, wx, v010.x * ax);
        const float c10y = __fmaf_rn(v110.y, wx, v010.y * ax);
        const float c11x = __fmaf_rn(v111.x, wx, v011.x * ax);
        const float c11y = __fmaf_rn(v111.y, wx, v011.y * ax);

        const float c0x = __fmaf_rn(c10x, wy, c00x * ay);
        const float c0y = __fmaf_rn(c10y, wy, c00y * ay);
        const float c1x = __fmaf_rn(c11x, wy, c01x * ay);
        const float c1y = __fmaf_rn(c11y, wy, c01y * ay);

        o[l].x = __fmaf_rn(c1x, wz, c0x * az);
        o[l].y = __fmaf_rn(c1y, wz, c0y * az);
    }

    // 32 contiguous floats per point -> 8 x non-temporal b128 stores.
    // NT keeps the 128MB output stream from evicting the L2-resident tables.
    v4f* __restrict__ op = (v4f*)(out + (size_t)p * (NLEV * 2));
#pragma unroll
    for (int q = 0; q < 8; ++q) {
        v4f val = { o[2 * q].x, o[2 * q].y, o[2 * q + 1].x, o[2 * q + 1].y };
        __builtin_nontemporal_store(val, op + q);
    }
}

extern "C" void kernel_launch(void* const* d_in, const int* in_sizes, int n_in,
                              void* d_out, int out_size, void* d_ws, size_t ws_size,
                              hipStream_t stream) {
    const float* x      = (const float*)d_in[0];   // (N, 3) float32
    const float* tables = (const float*)d_in[1];   // (16, 2^19, 2) float32
    float* out          = (float*)d_out;           // (N, 32) float32
    (void)d_ws; (void)ws_size; (void)n_in; (void)out_size;

    const int npts = in_sizes[0] / 3;

    // Resolutions computed exactly as the numpy reference (float64 path),
    // then halved (exact) so the kernel multiplies instead of divides.
    HalfRes hr;
    const double b = exp((log(512.0) - log(16.0)) / 15.0);
    for (int i = 0; i < NLEV; ++i)
        hr.h[i] = (float)floor(16.0 * pow(b, (double)i)) * 0.5f;

    const int blocks = (npts + BLOCK - 1) / BLOCK;
    hipLaunchKernelGGL(hashgrid_embed_kernel, dim3(blocks), dim3(BLOCK), 0, stream,
                       x, tables, out, hr, npts);
}